// CascadingSinkCachOriginal_26980984553672
// MI455X (gfx1250) — compile-verified
//
#include <hip/hip_runtime.h>

// Cascading-sink KV-cache update + gather.
// 0 matmul FLOPs; ~512 MiB read + ~512 MiB written => pure HBM-roofline
// problem (~46 us at 23.3 TB/s on MI455X). Strategy: one thread per 16B of
// output x 8 elements spaced 2^22 apart, so d4/s/region classification is
// computed ONCE per thread (and is wave-uniform -> SALU), then 8 independent
// b128 non-temporal loads followed by 8 non-temporal stores.

typedef float f4  __attribute__((ext_vector_type(4)));
typedef int   v4i __attribute__((ext_vector_type(4)));

// ---- gfx1250 async-to-LDS path (probe-verified: lowers to
// global_load_async_to_lds_b128 + s_wait_asynccnt). Kept as an unlaunched
// artifact; for a 1:1 streaming copy the LDS round-trip + coarse ASYNCcnt
// waits pipeline worse than 8-deep VGPR load batches. ----
#if defined(__has_builtin)
#if __has_builtin(__builtin_amdgcn_global_load_async_to_lds_b128) && \
    __has_builtin(__builtin_amdgcn_s_wait_asynccnt)
#define CDNA5_HAS_ASYNC_LDS 1
#endif
#endif
#ifndef CDNA5_HAS_ASYNC_LDS
#define CDNA5_HAS_ASYNC_LDS 0
#endif

#if CDNA5_HAS_ASYNC_LDS
__global__ void cdna5_async_lds_probe(const float* __restrict__ g,
                                      float* __restrict__ o) {
    __shared__ __attribute__((aligned(16))) float lds[256 * 4];
    auto gsrc = (__attribute__((address_space(1))) v4i*)(g + threadIdx.x * 4);
    auto ldst = (__attribute__((address_space(3))) v4i*)(lds + threadIdx.x * 4);
    __builtin_amdgcn_global_load_async_to_lds_b128(gsrc, ldst, 0, 0);
    __builtin_amdgcn_s_wait_asynccnt(0);
    __syncthreads();
    o[threadIdx.x] = lds[threadIdx.x * 4];
}
#endif

// Constants from the reference:
// B=2, H=32, D=128, W=1024, WL=1020, NC7=7, NUM_SINK=4, S_TOTAL=8192
// DO_CACHE=(T,T,T,F,...): chunks 0..2 shift left one token, chunk 3
// conditionally replaces its last slot, chunks 4..6 pass through.
// Output: [kv(2)][b(2)][h(32)][s(8192)][d(128)], s = sink(4)|last(1020)|chunks 6..0.
//
// Element k of a thread is flat float4-index v + k*2^22. Since 2^22 >> 18 = 16:
//   d4 = v&31 and s = (v>>5)&8191 are IDENTICAL for all k;
//   bhk = (v>>18) + 16k  ->  kv = k>>2, b = (k>>1)&1, h = (v>>18) + 16*(k&1).
// So the region/row decision is made once; only (kv,b,h) bases vary per k.

#define NTHREADS (1 << 22)

__global__ __launch_bounds__(256) void cascade_kv_gather(
    const float* __restrict__ key_states,    // (B,H,1,D)
    const float* __restrict__ value_states,  // (B,H,1,D)
    const float* __restrict__ sink_keys,     // (B,H,4,D)
    const float* __restrict__ sink_values,   // (B,H,4,D)
    const float* __restrict__ cache_keys,    // (7,B,H,1024,D)
    const float* __restrict__ cache_values,  // (7,B,H,1024,D)
    const float* __restrict__ cache_scores,  // (7,B,1024)
    const float* __restrict__ last_keys,     // (B,H,1020,D)
    const float* __restrict__ last_values,   // (B,H,1020,D)
    const float* __restrict__ attn_scores,   // (B,8192)
    float* __restrict__ out)                 // (2,B,H,8192,D)
{
    const int v    = blockIdx.x * 256 + threadIdx.x;  // < 2^22
    const int d4   = v & 31;          // float4 column within D=128 (per-lane)
    const int s    = (v >> 5) & 8191; // seq position (wave-uniform)
    const int bhk0 = v >> 18;         // 0..15     (wave-uniform)

    const float* keyv[2]   = {key_states,   value_states};
    const float* sinkv[2]  = {sink_keys,    sink_values};
    const float* cachev[2] = {cache_keys,   cache_values};
    const float* lastv[2]  = {last_keys,    last_values};

    f4 vals[8];

    if (s < 4) {
        // ---- sink region ----
#pragma unroll
        for (int k = 0; k < 8; ++k) {
            const int kv = k >> 2, b = (k >> 1) & 1, h = bhk0 + ((k & 1) << 4);
            vals[k] = __builtin_nontemporal_load(
                (const f4*)(sinkv[kv] + ((b * 32 + h) * 4 + s) * 128) + d4);
        }
    } else if (s < 1024) {
        // ---- last-window region (1020 rows) ----
        const int ls = s - 4;
#pragma unroll
        for (int k = 0; k < 8; ++k) {
            const int kv = k >> 2, b = (k >> 1) & 1, h = bhk0 + ((k & 1) << 4);
            vals[k] = __builtin_nontemporal_load(
                (const f4*)(lastv[kv] + ((b * 32 + h) * 1020 + ls) * 128) + d4);
        }
    } else {
        // ---- reversed cache chunks: out chunk c holds cache chunk i = 6-c ----
        const int j = s - 1024;
        const int w = j & 1023;
        const int i = 6 - (j >> 10);

        // Uniform replace decision (reference compares batch 0 only, then
        // broadcasts). Unfused mul+add to match XLA's beta*s + (1-beta)*a.
        // Only evaluated on the (i==3, w==1023) path.
        // Resolve (source chunk ci, source row wi) once; special-case the
        // cascaded new-token row.
        int ci = i, wi = w;
        bool use_new_token = false;
        if (i == 3) {
            if (w == 1023) {
                const float sc3 = __fadd_rn(__fmul_rn(0.99f, cache_scores[(3 * 2 + 0) * 1024 + 1023]),
                                            __fmul_rn(0.01f, attn_scores[4 + 3 * 1024 + 1023]));
                const float sev = __fadd_rn(__fmul_rn(0.99f, cache_scores[(2 * 2 + 0) * 1024 + 0]),
                                            __fmul_rn(0.01f, attn_scores[4 + 2 * 1024 + 0]));
                if (sc3 < sev) { ci = 2; wi = 0; }        // replace with chunk2's evicted head
                // else keep chunk3 row 1023
            }
        } else if (i <= 2) {
            if (w != 1023)      { wi = w + 1; }           // shifted left by one
            else if (i == 0)    { use_new_token = true; } // tail <- new K/V token
            else                { ci = i - 1; wi = 0; }   // tail <- chunk i-1's evicted head
        }
        // i in {4,5,6}: straight copy (ci=i, wi=w)

        if (use_new_token) {
#pragma unroll
            for (int k = 0; k < 8; ++k) {
                const int kv = k >> 2, b = (k >> 1) & 1, h = bhk0 + ((k & 1) << 4);
                vals[k] = __builtin_nontemporal_load(
                    (const f4*)(keyv[kv] + (b * 32 + h) * 128) + d4);
            }
        } else {
#pragma unroll
            for (int k = 0; k < 8; ++k) {
                const int kv = k >> 2, b = (k >> 1) & 1, h = bhk0 + ((k & 1) << 4);
                vals[k] = __builtin_nontemporal_load(
                    (const f4*)(cachev[kv] + (((ci * 2 + b) * 32 + h) * 1024 + wi) * 128) + d4);
            }
        }
    }

    // ---- 8 non-temporal b128 stores (working set >> 192MB L2) ----
    f4* ob = (f4*)out + v;
#pragma unroll
    for (int k = 0; k < 8; ++k)
        __builtin_nontemporal_store(vals[k], ob + k * NTHREADS);
}

extern "C" void kernel_launch(void* const* d_in, const int* in_sizes, int n_in,
                              void* d_out, int out_size, void* d_ws, size_t ws_size,
                              hipStream_t stream) {
    const float* key_states   = (const float*)d_in[0];
    const float* value_states = (const float*)d_in[1];
    const float* sink_keys    = (const float*)d_in[2];
    const float* sink_values  = (const float*)d_in[3];
    const float* cache_keys   = (const float*)d_in[4];
    const float* cache_values = (const float*)d_in[5];
    const float* cache_scores = (const float*)d_in[6];
    const float* last_keys    = (const float*)d_in[7];
    const float* last_values  = (const float*)d_in[8];
    // d_in[9] = last_scores: dead w.r.t. the output (reference updates it but
    // never returns it) — intentionally unused.
    const float* attn_scores  = (const float*)d_in[10];

    // out_size = 2*2*32*8192*128 = 2^27 floats = 2^25 float4s; 2^22 threads
    // each move exactly 8 float4s.
    const int threads = 256;               // 8 wave32 waves per workgroup
    const int blocks  = NTHREADS / 256;    // 16384

    cascade_kv_gather<<<blocks, threads, 0, stream>>>(
        key_states, value_states, sink_keys, sink_values,
        cache_keys, cache_values, cache_scores,
        last_keys, last_values, attn_scores,
        (float*)d_out);
}